// PreProcesser_84670985273723
// MI455X (gfx1250) — compile-verified
//
#include <hip/hip_runtime.h>
#include <math.h>

// MI455X / gfx1250, wave32. One block per residue (ZN=1024 blocks, 256 thr = 8 waves).
// Field phase: fp32 VALU (sqrt+div per voxel-atom, matches JAX fp32 semantics).
// Divergence phase: exact 16x16 difference-matrix matmuls via V_WMMA_F32_16X16X4_F32.

typedef float v2f __attribute__((ext_vector_type(2)));
typedef float v8f __attribute__((ext_vector_type(8)));

#define VDIM 16
#define NVOX 4096   // 16^3
#define NATOM 14
#define NLBL 21

// Central-difference matrix (CELL_DIM = 1):
// row 0: f[1]-f[0]; rows 1..14: (f[i+1]-f[i-1])/2; row 15: f[15]-f[14]
__device__ __forceinline__ float dm_elem(int m, int k) {
  if (m == 0)  return (k == 0)  ? -1.0f : ((k == 1)  ? 1.0f : 0.0f);
  if (m == 15) return (k == 15) ?  1.0f : ((k == 14) ? -1.0f : 0.0f);
  if (k == m - 1) return -0.5f;
  if (k == m + 1) return  0.5f;
  return 0.0f;
}

// One derivative pass: out(+)= Dm x src along one axis, 16 column-tiles of 16.
// PASS 0: dx[x,y,z] = sum_k Dm[x,k] fx[k,y,z]   (m=x, tile t=y, n=z), overwrite dst
// PASS 1: dy[x,y,z] = sum_k Dm[y,k] fy[x,k,z]   (m=y, tile t=x, n=z), accumulate
// PASS 2: dz[x,y,z] = sum_k Dm[z,k] fz[x,y,k]   (m=z, tile t=x, n=y), accumulate
template <int PASS>
__device__ __forceinline__ void div_pass(const float* __restrict__ src,
                                         float* __restrict__ dst,
                                         const v2f aReg[4], int lane, int wv) {
  const int nn    = lane & 15;
  const int khalf = (lane < 16) ? 0 : 2;   // A/B layout: VGPR0 holds K, K+2 halves
  const int mbase = (lane < 16) ? 0 : 8;   // D layout: VGPR j holds rows j and j+8
  for (int t = wv; t < 16; t += 8) {       // 16 tiles over 8 waves
    v8f acc = {};
#pragma unroll
    for (int kk = 0; kk < 4; ++kk) {       // K = 16 as 4 steps of K=4
      const int k0 = 4 * kk + khalf;
      const int k1 = k0 + 1;
      int i0, i1;
      if (PASS == 0)      { i0 = k0 * 256 + t * 16 + nn; i1 = k1 * 256 + t * 16 + nn; }
      else if (PASS == 1) { i0 = t * 256 + k0 * 16 + nn; i1 = t * 256 + k1 * 16 + nn; }
      else                { i0 = t * 256 + nn * 16 + k0; i1 = t * 256 + nn * 16 + k1; }
      v2f b; b.x = src[i0]; b.y = src[i1];
      acc = __builtin_amdgcn_wmma_f32_16x16x4_f32(
          /*neg_a=*/false, aReg[kk], /*neg_b=*/false, b,
          /*c_mod=*/(short)0, acc, /*reuse_a=*/false, /*reuse_b=*/false);
    }
#pragma unroll
    for (int j = 0; j < 8; ++j) {
      const int m = mbase + j;
      int idx;
      if (PASS == 0)      idx = m * 256 + t * 16 + nn;
      else if (PASS == 1) idx = t * 256 + m * 16 + nn;
      else                idx = t * 256 + nn * 16 + m;
      if (PASS == 0) dst[idx] = acc[j];
      else           dst[idx] += acc[j];
    }
  }
}

__global__ __launch_bounds__(256)
void voxel_field_div_84670985273723(const float* __restrict__ C,
                                    const int*   __restrict__ L,
                                    const float* __restrict__ atom_mask,
                                    const float* __restrict__ charges,
                                    float* __restrict__ out, int ZN) {
  __shared__ float sFx[NVOX], sFy[NVOX], sFz[NVOX];          // 48 KB
  __shared__ float sF[9], sO[3];
  __shared__ float sAx[NATOM], sAy[NATOM], sAz[NATOM], sPc[NATOM];

  const int n    = blockIdx.x;
  const int tid  = threadIdx.x;
  const int lane = tid & 31;
  const int wv   = tid >> 5;

  float* outCb  = out;                       // (ZN,4,3)
  float* outDiv = out + (size_t)ZN * 12;     // (ZN,1,16,16,16)
  float* outFr  = outDiv + (size_t)ZN * NVOX; // (ZN,3,3)

  // Constant difference-matrix A operand, per-lane, built once (16x4 f32 layout).
  v2f aReg[4];
  {
    const int m     = lane & 15;
    const int khalf = (lane < 16) ? 0 : 2;
#pragma unroll
    for (int kk = 0; kk < 4; ++kk) {
      aReg[kk].x = dm_elem(m, 4 * kk + khalf);
      aReg[kk].y = dm_elem(m, 4 * kk + khalf + 1);
    }
  }

  // ---- Phase 1: atoms, point charges, backbone, frames ----
  if (tid < NATOM) {
    const float* cn = C + (size_t)n * (NATOM * 3);
    sAx[tid] = cn[tid * 3 + 0];
    sAy[tid] = cn[tid * 3 + 1];
    sAz[tid] = cn[tid * 3 + 2];
    int lbl = L[n];
    if (lbl == -1) lbl = NLBL - 1;
    sPc[tid] = charges[lbl * NATOM + tid] * atom_mask[(size_t)n * NATOM + tid];
  }
  if (tid == 0) {
    const float* cn = C + (size_t)n * (NATOM * 3);
    const float Nx = cn[0], Ny = cn[1], Nz = cn[2];
    const float Ax = cn[3], Ay = cn[4], Az = cn[5];   // CA
    const float Cx = cn[6], Cy = cn[7], Cz = cn[8];   // C
    const float b1x = Ax - Nx, b1y = Ay - Ny, b1z = Az - Nz;
    const float b2x = Cx - Ax, b2y = Cy - Ay, b2z = Cz - Az;
    const float b3x = b1y * b2z - b1z * b2y;
    const float b3y = b1z * b2x - b1x * b2z;
    const float b3z = b1x * b2y - b1y * b2x;
    const float cbx = Ax - 0.58273431f * b2x + 0.56802827f * b1x - 0.54067466f * b3x;
    const float cby = Ay - 0.58273431f * b2y + 0.56802827f * b1y - 0.54067466f * b3y;
    const float cbz = Az - 0.58273431f * b2z + 0.56802827f * b1z - 0.54067466f * b3z;

    float* cb_out = outCb + (size_t)n * 12;
    cb_out[0] = Nx;  cb_out[1]  = Ny;  cb_out[2]  = Nz;
    cb_out[3] = Ax;  cb_out[4]  = Ay;  cb_out[5]  = Az;
    cb_out[6] = Cx;  cb_out[7]  = Cy;  cb_out[8]  = Cz;
    cb_out[9] = cbx; cb_out[10] = cby; cb_out[11] = cbz;

    // frames (replicates reference exactly, incl. scalar broadcast subtraction)
    float yx = cbx - Ax, yy = cby - Ay, yz = cbz - Az;
    float yn = fmaxf(sqrtf(yx * yx + yy * yy + yz * yz), 1e-6f);
    yx /= yn; yy /= yn; yz /= yn;
    float xrx = Cx - Nx, xry = Cy - Ny, xrz = Cz - Nz;
    const float xp = xrx * yx + xry * yy + xrz * yz;  // scalar
    float xx = xrx - xp, xy = xry - xp, xz = xrz - xp;
    float xn = fmaxf(sqrtf(xx * xx + xy * xy + xz * xz), 1e-6f);
    xx /= xn; xy /= xn; xz /= xn;
    float zx = xy * yz - xz * yy;
    float zy = xz * yx - xx * yz;
    float zz = xx * yy - xy * yx;
    float zn = fmaxf(sqrtf(zx * zx + zy * zy + zz * zz), 1e-6f);
    zx /= zn; zy /= zn; zz /= zn;

    sF[0] = xx; sF[1] = xy; sF[2] = xz;
    sF[3] = yx; sF[4] = yy; sF[5] = yz;
    sF[6] = zx; sF[7] = zy; sF[8] = zz;
    sO[0] = cbx; sO[1] = cby; sO[2] = cbz;

    float* fr = outFr + (size_t)n * 9;
    fr[0] = xx; fr[1] = xy; fr[2] = xz;
    fr[3] = yx; fr[4] = yy; fr[5] = yz;
    fr[6] = zx; fr[7] = zy; fr[8] = zz;
  }
  __syncthreads();

  // ---- Phase 2: normalized inverse-square field on the 16^3 grid ----
  for (int i = tid; i < NVOX; i += 256) {
    const int x = i >> 8, y = (i >> 4) & 15, z = i & 15;
    const float vx = (float)x - 8.0f;
    const float vy = (float)y - 4.0f;
    const float vz = (float)z - 8.0f;
    const float px = sO[0] + vx * sF[0] + vy * sF[3] + vz * sF[6];
    const float py = sO[1] + vx * sF[1] + vy * sF[4] + vz * sF[7];
    const float pz = sO[2] + vx * sF[2] + vy * sF[5] + vz * sF[8];
    float fx = 0.0f, fy = 0.0f, fz = 0.0f;
#pragma unroll
    for (int a = 0; a < NATOM; ++a) {
      const float dx = px - sAx[a];
      const float dy = py - sAy[a];
      const float dz = pz - sAz[a];
      const float dd = dx * dx + dy * dy + dz * dz;
      const float d  = sqrtf(dd);
      const float dc = fmaxf(d, 1.0f);             // clip(d, CELL_DIM)
      const float ds = (d == 0.0f) ? 1.0f : d;     // where(d==0, 1, d)
      const float w  = sPc[a] / (ds * dc * dc);
      fx += w * dx; fy += w * dy; fz += w * dz;
    }
    const float fn = sqrtf(fx * fx + fy * fy + fz * fz);
    const float inv = (fn == 0.0f) ? 1.0f : 1.0f / fn;
    sFx[i] = fx * inv; sFy[i] = fy * inv; sFz[i] = fz * inv;
  }
  __syncthreads();

  // ---- Phase 3: divergence as 16x16 difference-matrix WMMAs ----
  // Pass 0 overwrites sFx in place (each wave writes exactly the columns it read).
  div_pass<0>(sFx, sFx, aReg, lane, wv);
  __syncthreads();
  div_pass<1>(sFy, sFx, aReg, lane, wv);
  __syncthreads();
  div_pass<2>(sFz, sFx, aReg, lane, wv);
  __syncthreads();

  // ---- Phase 4: write div ----
  float* dv = outDiv + (size_t)n * NVOX;
  for (int i = tid; i < NVOX; i += 256) dv[i] = sFx[i];
}

extern "C" void kernel_launch(void* const* d_in, const int* in_sizes, int n_in,
                              void* d_out, int out_size, void* d_ws, size_t ws_size,
                              hipStream_t stream) {
  const float* C         = (const float*)d_in[0];
  const int*   L         = (const int*)d_in[1];
  const float* atom_mask = (const float*)d_in[2];
  const float* charges   = (const float*)d_in[3];
  float* out = (float*)d_out;
  const int ZN = in_sizes[0] / (NATOM * 3);
  voxel_field_div_84670985273723<<<ZN, 256, 0, stream>>>(C, L, atom_mask, charges, out, ZN);
}